// DeepIM_41996190220665
// MI455X (gfx1250) — compile-verified
//
#include <hip/hip_runtime.h>
#include <hip/hip_bf16.h>
#include <math.h>

typedef float v2f __attribute__((ext_vector_type(2)));
typedef float v8f __attribute__((ext_vector_type(8)));

#define N_NODES 4096
#define LRELU_SLOPE 0.2f

// ---------------------------------------------------------------------------
// GEMV core on V_WMMA_F32_16X16X4_F32.
// Block = 256 threads (8 waves). Each block produces 16 outputs (one M-tile).
// Each wave takes a K/8 slice; A = 16 rows of W x 4 k, B = x broadcast across
// all 16 columns, so every column of D equals the 16 partial outputs.
// A layout (32-bit 16x4): lanes 0-15 -> K=0,1 in v[0],v[1]; lanes 16-31 -> K=2,3.
// B layout mirrors the K split; since B columns are replicated, only the K
// placement matters. Partial sums are combined across waves via LDS.
// ---------------------------------------------------------------------------
static __device__ __forceinline__ float gemv16_wmma(
    const float* __restrict__ W, const float* __restrict__ x,
    int K, int mtile, float* lds)
{
    const int tid  = threadIdx.x;
    const int lane = tid & 31;
    const int wv   = tid >> 5;            // wave id 0..7
    const int ksl  = K >> 3;              // per-wave K slice (K multiple of 64)
    const int k0   = wv * ksl;
    const int kh   = (lane >> 4) << 1;    // 0 for lanes 0-15, 2 for lanes 16-31
    const int row  = (mtile << 4) + (lane & 15);

    const float* wp = W + (size_t)row * (size_t)K + k0 + kh;
    const float* xp = x + k0 + kh;

    v8f acc0 = {}; v8f acc1 = {};
    for (int kb = 0; kb < ksl; kb += 8) {
        v2f a0 = *(const v2f*)(wp + kb);
        v2f b0 = { xp[kb], xp[kb + 1] };
        acc0 = __builtin_amdgcn_wmma_f32_16x16x4_f32(
            false, a0, false, b0, (short)0, acc0, false, false);
        v2f a1 = *(const v2f*)(wp + kb + 4);
        v2f b1 = { xp[kb + 4], xp[kb + 5] };
        acc1 = __builtin_amdgcn_wmma_f32_16x16x4_f32(
            false, a1, false, b1, (short)0, acc1, false, false);
    }
    v8f acc = acc0 + acc1;

    // D layout: VGPR r, lanes 0-15 hold M=r; lanes 16-31 hold M=r+8.
    // All columns identical -> lanes 0 and 16 publish 8 rows each.
    if ((lane & 15) == 0) {
        const int base = wv * 16 + (lane >> 4) * 8;
        #pragma unroll
        for (int r = 0; r < 8; ++r) lds[base + r] = acc[r];
    }
    __syncthreads();
    float v = 0.0f;
    if (tid < 16) {
        #pragma unroll
        for (int w = 0; w < 8; ++w) v += lds[w * 16 + tid];
    }
    __syncthreads();   // make lds reusable by a following call
    return v;
}

// ACT: 0 = relu, 1 = none, 2 = sigmoid
template<int ACT>
__global__ __launch_bounds__(256) void gemv_act_kernel(
    const float* __restrict__ W, const float* __restrict__ x,
    const float* __restrict__ bias, float* __restrict__ out, int K)
{
    __shared__ float lds[128];
    float v = gemv16_wmma(W, x, K, blockIdx.x, lds);
    if (threadIdx.x < 16) {
        const int m = (blockIdx.x << 4) + threadIdx.x;
        v += bias[m];
        if (ACT == 0)      v = fmaxf(v, 0.0f);
        else if (ACT == 2) v = 1.0f / (1.0f + __expf(-v));
        out[m] = v;
    }
}

// mu / logvar / z in one pass (two WMMA GEMVs per tile)
__global__ __launch_bounds__(256) void vae_head_kernel(
    const float* __restrict__ w3, const float* __restrict__ b3,
    const float* __restrict__ w4, const float* __restrict__ b4,
    const float* __restrict__ h,  const float* __restrict__ eps,
    float* __restrict__ mu, float* __restrict__ logvar, float* __restrict__ z,
    int K)
{
    __shared__ float lds[128];
    float mv  = gemv16_wmma(w3, h, K, blockIdx.x, lds);
    float lvv = gemv16_wmma(w4, h, K, blockIdx.x, lds);
    if (threadIdx.x < 16) {
        const int i = (blockIdx.x << 4) + threadIdx.x;
        mv  += b3[i];
        lvv += b4[i];
        mu[i] = mv;
        logvar[i] = lvv;
        z[i] = mv + eps[i] * __expf(0.5f * lvv);
    }
}

// Per-head scalars: alpha_h=W_h.a[:64], beta_h=W_h.a[64:], gamma_h=W_h.outW_h;
// plus sum(x_hat) for the empty-row softmax fallback.
__global__ __launch_bounds__(256) void gat_prep_kernel(
    const float* __restrict__ gatW, const float* __restrict__ gata,
    const float* __restrict__ outW, const float* __restrict__ xhat,
    float* __restrict__ scal)
{
    __shared__ float lds[256];
    const int tid = threadIdx.x, wv = tid >> 5, lane = tid & 31;
    if (wv < 4) {
        const int h = wv;
        const float* W  = gatW + h * 64;
        const float* a  = gata + h * 128;
        const float* oW = outW + h * 64;
        const float w0 = W[lane], w1 = W[lane + 32];
        float pa = w0 * a[lane]      + w1 * a[lane + 32];
        float pb = w0 * a[64 + lane] + w1 * a[96 + lane];
        float pg = w0 * oW[lane]     + w1 * oW[lane + 32];
        #pragma unroll
        for (int off = 16; off > 0; off >>= 1) {
            pa += __shfl_xor(pa, off, 32);
            pb += __shfl_xor(pb, off, 32);
            pg += __shfl_xor(pg, off, 32);
        }
        if (lane == 0) { scal[h] = pa; scal[4 + h] = pb; scal[8 + h] = pg; }
    }
    float s = 0.0f;
    for (int i = tid; i < N_NODES; i += 256) s += xhat[i];
    lds[tid] = s; __syncthreads();
    for (int st = 128; st > 0; st >>= 1) {
        if (tid < st) lds[tid] += lds[tid + st];
        __syncthreads();
    }
    if (tid == 0) scal[12] = lds[0];
}

__global__ __launch_bounds__(256) void reduce_sum_kernel(
    const float* __restrict__ v, int n, float* __restrict__ out)
{
    __shared__ float lds[256];
    const int tid = threadIdx.x;
    float s = 0.0f;
    for (int i = tid; i < n; i += 256) s += v[i];
    lds[tid] = s; __syncthreads();
    for (int st = 128; st > 0; st >>= 1) {
        if (tid < st) lds[tid] += lds[tid + st];
        __syncthreads();
    }
    if (tid == 0) *out = lds[0];
}

// GAT layer 1 (4 heads fused) -> Wh2[i] = sum_h gamma_h * c_h[i].
// Online masked softmax, one wave per row. Masked entries contribute exactly 0
// (exp(-9e15 - m) flushes to 0 in f32); empty rows -> uniform -> mean(x_hat).
__global__ __launch_bounds__(256) void gat1_kernel(
    const int* __restrict__ adj, const float* __restrict__ xhat,
    const float* __restrict__ scal, float* __restrict__ wh2)
{
    const int row  = (blockIdx.x << 3) + (threadIdx.x >> 5);
    const int lane = threadIdx.x & 31;
    const float xi = xhat[row];

    float al[4], be[4], ga[4], m[4], l[4], s[4];
    #pragma unroll
    for (int h = 0; h < 4; ++h) {
        al[h] = scal[h]; be[h] = scal[4 + h]; ga[h] = scal[8 + h];
        m[h] = -INFINITY; l[h] = 0.0f; s[h] = 0.0f;
    }

    const int* arow = adj + (size_t)row * N_NODES;
    for (int j = lane; j < N_NODES; j += 32) {
        if (arow[j] != 0) {
            const float xj = xhat[j];
            #pragma unroll
            for (int h = 0; h < 4; ++h) {
                float e = al[h] * xi + be[h] * xj;
                e = (e >= 0.0f) ? e : LRELU_SLOPE * e;
                const float nm = fmaxf(m[h], e);
                const float sc = __expf(m[h] - nm);   // -inf - finite -> 0
                const float w  = __expf(e - nm);
                l[h] = l[h] * sc + w;
                s[h] = s[h] * sc + w * xj;
                m[h] = nm;
            }
        }
    }

    #pragma unroll
    for (int off = 16; off > 0; off >>= 1) {
        #pragma unroll
        for (int h = 0; h < 4; ++h) {
            const float mo = __shfl_xor(m[h], off, 32);
            const float lo = __shfl_xor(l[h], off, 32);
            const float so = __shfl_xor(s[h], off, 32);
            const float nm = fmaxf(m[h], mo);
            const float sa = (l[h] > 0.0f) ? __expf(m[h] - nm) : 0.0f;
            const float sb = (lo   > 0.0f) ? __expf(mo   - nm) : 0.0f;
            l[h] = l[h] * sa + lo * sb;
            s[h] = s[h] * sa + so * sb;
            m[h] = nm;
        }
    }

    if (lane == 0) {
        const float cmean = scal[12] * (1.0f / (float)N_NODES);
        float acc = 0.0f;
        #pragma unroll
        for (int h = 0; h < 4; ++h) {
            const float c = (l[h] > 0.0f) ? (s[h] / l[h]) : cmean;
            acc += ga[h] * c;
        }
        wh2[row] = acc;
    }
}

// GAT layer 2 (single head on Wh2) + ELU -> y_hat
__global__ __launch_bounds__(256) void gat2_kernel(
    const int* __restrict__ adj, const float* __restrict__ wh2,
    const float* __restrict__ out_a, const float* __restrict__ sum_wh2,
    float* __restrict__ yhat)
{
    const int row  = (blockIdx.x << 3) + (threadIdx.x >> 5);
    const int lane = threadIdx.x & 31;
    const float a0 = out_a[0], a1 = out_a[1];
    const float wi = wh2[row];

    float m = -INFINITY, l = 0.0f, s = 0.0f;
    const int* arow = adj + (size_t)row * N_NODES;
    for (int j = lane; j < N_NODES; j += 32) {
        if (arow[j] != 0) {
            const float wj = wh2[j];
            float e = a0 * wi + a1 * wj;
            e = (e >= 0.0f) ? e : LRELU_SLOPE * e;
            const float nm = fmaxf(m, e);
            const float sc = __expf(m - nm);
            const float w  = __expf(e - nm);
            l = l * sc + w;
            s = s * sc + w * wj;
            m = nm;
        }
    }
    #pragma unroll
    for (int off = 16; off > 0; off >>= 1) {
        const float mo = __shfl_xor(m, off, 32);
        const float lo = __shfl_xor(l, off, 32);
        const float so = __shfl_xor(s, off, 32);
        const float nm = fmaxf(m, mo);
        const float sa = (l  > 0.0f) ? __expf(m  - nm) : 0.0f;
        const float sb = (lo > 0.0f) ? __expf(mo - nm) : 0.0f;
        l = l * sa + lo * sb;
        s = s * sa + so * sb;
        m = nm;
    }
    if (lane == 0) {
        const float o = (l > 0.0f) ? (s / l) : (*sum_wh2) * (1.0f / (float)N_NODES);
        yhat[row] = (o > 0.0f) ? o : (__expf(o) - 1.0f);   // ELU, alpha=1
    }
}

extern "C" void kernel_launch(void* const* d_in, const int* in_sizes, int n_in,
                              void* d_out, int out_size, void* d_ws, size_t ws_size,
                              hipStream_t stream)
{
    const float* x      = (const float*)d_in[0];
    const float* eps    = (const float*)d_in[1];
    const int*   adj    = (const int*)  d_in[2];
    const float* enc_w1 = (const float*)d_in[3];
    const float* enc_b1 = (const float*)d_in[4];
    const float* enc_w2 = (const float*)d_in[5];
    const float* enc_b2 = (const float*)d_in[6];
    const float* enc_w3 = (const float*)d_in[7];
    const float* enc_b3 = (const float*)d_in[8];
    const float* enc_w4 = (const float*)d_in[9];
    const float* enc_b4 = (const float*)d_in[10];
    const float* dec_w1 = (const float*)d_in[11];
    const float* dec_b1 = (const float*)d_in[12];
    const float* dec_w2 = (const float*)d_in[13];
    const float* dec_b2 = (const float*)d_in[14];
    const float* dec_w3 = (const float*)d_in[15];
    const float* dec_b3 = (const float*)d_in[16];
    const float* gat_W  = (const float*)d_in[17];
    const float* gat_a  = (const float*)d_in[18];
    const float* out_W  = (const float*)d_in[19];
    const float* out_a  = (const float*)d_in[20];

    float* out   = (float*)d_out;
    float* xhat  = out;            // [0, 4096)
    float* yhat  = out + 4096;     // [4096, 8192)
    float* mu    = out + 8192;     // [8192, 8704)
    float* lvar  = out + 8704;     // [8704, 9216)

    float* ws   = (float*)d_ws;
    float* h1   = ws;              // 1024
    float* h2   = ws + 1024;       // 1024
    float* z    = ws + 2048;       // 512
    float* h3   = ws + 2560;       // 1024
    float* h4   = ws + 3584;       // 1024
    float* wh2  = ws + 4608;       // 4096
    float* scal = ws + 8704;       // [0..3]=alpha [4..7]=beta [8..11]=gamma [12]=sum_xhat [13]=sum_wh2

    const dim3 blk(256);

    // Encoder
    gemv_act_kernel<0><<<dim3(64),  blk, 0, stream>>>(enc_w1, x,  enc_b1, h1, 4096);
    gemv_act_kernel<0><<<dim3(64),  blk, 0, stream>>>(enc_w2, h1, enc_b2, h2, 1024);
    vae_head_kernel   <<<dim3(32),  blk, 0, stream>>>(enc_w3, enc_b3, enc_w4, enc_b4,
                                                      h2, eps, mu, lvar, z, 1024);
    // Decoder
    gemv_act_kernel<0><<<dim3(64),  blk, 0, stream>>>(dec_w1, z,  dec_b1, h3, 512);
    gemv_act_kernel<0><<<dim3(64),  blk, 0, stream>>>(dec_w2, h3, dec_b2, h4, 1024);
    gemv_act_kernel<2><<<dim3(256), blk, 0, stream>>>(dec_w3, h4, dec_b3, xhat, 1024);

    // GAT
    gat_prep_kernel   <<<dim3(1),   blk, 0, stream>>>(gat_W, gat_a, out_W, xhat, scal);
    gat1_kernel       <<<dim3(512), blk, 0, stream>>>(adj, xhat, scal, wh2);
    reduce_sum_kernel <<<dim3(1),   blk, 0, stream>>>(wh2, N_NODES, scal + 13);
    gat2_kernel       <<<dim3(512), blk, 0, stream>>>(adj, wh2, out_a, scal + 13, yhat);
}